// PDEFunc_24498493456985
// MI455X (gfx1250) — compile-verified
//
#include <hip/hip_runtime.h>
#include <hip/hip_bf16.h>

// ---------------------------------------------------------------------------
// Neural-ODE (RK4, NT=100) over 3 independent tiny MLPs, then basis expansion.
//
// Phase 1 (latency-bound): 3 blocks (one per field), 8 waves each. All weights
//   preloaded into per-wave registers as V_WMMA_F32_16X16X4_F32 B-fragments
//   (W2 tile = 32 x v2f = 64 VGPRs). Activations go through LDS with padded
//   row strides (34 / 132 floats) so A-fragment gathers are bank-conflict free.
//   M dim = batch (8 real rows padded to 16; pad rows stay finite, never read).
// Phase 2 (store-bound, ~157MB @ 23.3TB/s ≈ 7us): coeffs[3,800,32] x
//   basis[3,32,16384] as fp32 WMMA, one 16x16 tile per wave, basis hot in L2.
// ---------------------------------------------------------------------------

typedef __attribute__((ext_vector_type(2))) float v2f;
typedef __attribute__((ext_vector_type(8))) float v8f;

#define KD      32
#define HD      128
#define MBATCH  8
#define NTIME   100
#define NPIX    (128 * 128)
#define AS      34    // LDS row stride for [16][32] buffers (bank-conflict free)
#define HS      132   // LDS row stride for [16][128] buffers

static __device__ __forceinline__ v8f wmma4(v2f a, v2f b, v8f c) {
  // D = A(16x4,f32) * B(4x16,f32) + C(16x16,f32)
  return __builtin_amdgcn_wmma_f32_16x16x4_f32(
      /*neg_a=*/false, a, /*neg_b=*/false, b,
      /*c_mod=*/(short)0, c, /*reuse_a=*/false, /*reuse_b=*/false);
}

// ---------------------------------------------------------------------------
// Kernel 1: full RK4 trajectory for one field per block.
// coeffs layout: [3][NTIME][MBATCH][KD]
// ---------------------------------------------------------------------------
__launch_bounds__(256, 1)
__global__ void ode_rk4_kernel(const float* __restrict__ t,
                               const float* __restrict__ init_coeffs,
                               const float* __restrict__ W1, const float* __restrict__ b1,
                               const float* __restrict__ W2, const float* __restrict__ b2,
                               const float* __restrict__ W3, const float* __restrict__ b3,
                               float* __restrict__ coeffs) {
  __shared__ float sy [16 * AS];   // current state y
  __shared__ float syt[16 * AS];   // RK stage input
  __shared__ float ska[16 * AS];   // k accumulator (k1 + 2k2 + 2k3)
  __shared__ float sko[16 * AS];   // MLP output (current k_i)
  __shared__ float sh1[16 * HS];   // hidden 1
  __shared__ float sh2[16 * HS];   // hidden 2

  const int f    = blockIdx.x;
  const int tid  = threadIdx.x;
  const int wave = tid >> 5;
  const int lane = tid & 31;
  const int lh   = lane & 15;      // row/col within half-wave
  const int kh   = lane >> 4;      // which K/M half this lane covers

  const float dt  = (t[NTIME - 1] - t[0]) / (float)(NTIME - 1);
  const float hdt = 0.5f * dt;
  const float dt6 = dt / 6.0f;

  // ---- preload weight fragments into registers (WMMA B layout) ----
  const int n0  = wave * 16;       // N-tile base for this wave
  const int col = n0 + lh;
  v2f B1f[8], B2f[32], B3f[32];
  float bias1, bias2, bias3 = 0.0f;
  {
    const float* w1 = W1 + (size_t)f * HD * KD;   // [HD][KD]
    const float* w2 = W2 + (size_t)f * HD * HD;   // [HD][HD]
#pragma unroll
    for (int kk = 0; kk < 8; ++kk) {
      int kb = kk * 4 + 2 * kh;
      B1f[kk].x = w1[col * KD + kb];
      B1f[kk].y = w1[col * KD + kb + 1];
    }
#pragma unroll
    for (int kk = 0; kk < 32; ++kk) {
      int kb = kk * 4 + 2 * kh;
      B2f[kk].x = w2[col * HD + kb];
      B2f[kk].y = w2[col * HD + kb + 1];
    }
    bias1 = b1[f * HD + col];
    bias2 = b2[f * HD + col];
    if (wave < 2) {                               // layer-3 output is only 32 wide
      const float* w3 = W3 + (size_t)f * KD * HD; // [KD][HD]
#pragma unroll
      for (int kk = 0; kk < 32; ++kk) {
        int kb = kk * 4 + 2 * kh;
        B3f[kk].x = w3[col * HD + kb];
        B3f[kk].y = w3[col * HD + kb + 1];
      }
      bias3 = b3[f * KD + col];
    }
  }

  // ---- init state (pad rows 8..15 with zero so everything stays finite) ----
  for (int i = tid; i < 16 * KD; i += 256) {
    int r = i >> 5, k = i & 31;
    float v = (r < MBATCH) ? init_coeffs[f * KD + k] : 0.0f;
    sy [r * AS + k] = v;
    syt[r * AS + k] = v;
    ska[r * AS + k] = 0.0f;
  }
  for (int i = tid; i < MBATCH * KD; i += 256)    // coeffs[f][0]
    coeffs[((size_t)f * NTIME) * MBATCH * KD + i] = init_coeffs[f * KD + (i & 31)];
  __syncthreads();

  // ---- MLP evaluation: in -> sko  (ReLU, ELU, linear) ----
  auto mlp = [&](const float* __restrict__ in) {
    // layer 1: [16,32] x [32,16(tile)]
    v8f a0 = {}, a1 = {};
#pragma unroll
    for (int kk = 0; kk < 8; kk += 2) {
      int kb0 = kk * 4 + 2 * kh;
      v2f a; a.x = in[lh * AS + kb0]; a.y = in[lh * AS + kb0 + 1];
      a0 = wmma4(a, B1f[kk], a0);
      int kb1 = (kk + 1) * 4 + 2 * kh;
      v2f a2; a2.x = in[lh * AS + kb1]; a2.y = in[lh * AS + kb1 + 1];
      a1 = wmma4(a2, B1f[kk + 1], a1);
    }
    v8f acc = a0 + a1;
#pragma unroll
    for (int r = 0; r < 8; ++r) {
      float v = acc[r] + bias1;
      sh1[(r + 8 * kh) * HS + col] = v > 0.0f ? v : 0.0f;   // ReLU
    }
    __syncthreads();

    // layer 2: [16,128] x [128,16(tile)], ELU
    a0 = {}; a1 = {};
#pragma unroll
    for (int kk = 0; kk < 32; kk += 2) {
      int kb0 = kk * 4 + 2 * kh;
      v2f a; a.x = sh1[lh * HS + kb0]; a.y = sh1[lh * HS + kb0 + 1];
      a0 = wmma4(a, B2f[kk], a0);
      int kb1 = (kk + 1) * 4 + 2 * kh;
      v2f a2; a2.x = sh1[lh * HS + kb1]; a2.y = sh1[lh * HS + kb1 + 1];
      a1 = wmma4(a2, B2f[kk + 1], a1);
    }
    acc = a0 + a1;
#pragma unroll
    for (int r = 0; r < 8; ++r) {
      float v = acc[r] + bias2;
      sh2[(r + 8 * kh) * HS + col] = v > 0.0f ? v : (__expf(v) - 1.0f);  // ELU
    }
    __syncthreads();

    // layer 3: [16,128] x [128,16(tile)], only waves 0/1 (uniform branch, EXEC stays full)
    if (wave < 2) {
      a0 = {}; a1 = {};
#pragma unroll
      for (int kk = 0; kk < 32; kk += 2) {
        int kb0 = kk * 4 + 2 * kh;
        v2f a; a.x = sh2[lh * HS + kb0]; a.y = sh2[lh * HS + kb0 + 1];
        a0 = wmma4(a, B3f[kk], a0);
        int kb1 = (kk + 1) * 4 + 2 * kh;
        v2f a2; a2.x = sh2[lh * HS + kb1]; a2.y = sh2[lh * HS + kb1 + 1];
        a1 = wmma4(a2, B3f[kk + 1], a1);
      }
      acc = a0 + a1;
#pragma unroll
      for (int r = 0; r < 8; ++r)
        sko[(r + 8 * kh) * AS + col] = acc[r] + bias3;
    }
    __syncthreads();
  };

  // ---- RK4 time stepping ----
  for (int s = 0; s < NTIME - 1; ++s) {
    mlp(sy);                                      // k1 -> sko
    for (int i = tid; i < 16 * KD; i += 256) {
      int off = (i >> 5) * AS + (i & 31);
      float k1 = sko[off];
      ska[off] = k1;
      syt[off] = sy[off] + hdt * k1;
    }
    __syncthreads();

    mlp(syt);                                     // k2
    for (int i = tid; i < 16 * KD; i += 256) {
      int off = (i >> 5) * AS + (i & 31);
      float k2 = sko[off];
      ska[off] += 2.0f * k2;
      syt[off] = sy[off] + hdt * k2;
    }
    __syncthreads();

    mlp(syt);                                     // k3
    for (int i = tid; i < 16 * KD; i += 256) {
      int off = (i >> 5) * AS + (i & 31);
      float k3 = sko[off];
      ska[off] += 2.0f * k3;
      syt[off] = sy[off] + dt * k3;
    }
    __syncthreads();

    mlp(syt);                                     // k4
    for (int i = tid; i < 16 * KD; i += 256) {
      int off = (i >> 5) * AS + (i & 31);
      sy[off] += dt6 * (ska[off] + sko[off]);
    }
    __syncthreads();

    float* dst = coeffs + (((size_t)f * NTIME) + (s + 1)) * MBATCH * KD;
    for (int i = tid; i < MBATCH * KD; i += 256)
      dst[i] = sy[(i >> 5) * AS + (i & 31)];
    // no barrier needed: next iteration only *reads* sy and writes other buffers
  }
}

// ---------------------------------------------------------------------------
// Kernel 2: soln[t,b,f,x,y] = sum_k coeffs[f,t,b,k] * basis[f,k,x,y]
// One 16(M=t*8+b) x 16(N=pixel) WMMA tile per wave; 8 waves/block cover 128
// pixels. Store-bound: basis (6MB) stays L2-resident, output streams to HBM.
// ---------------------------------------------------------------------------
__launch_bounds__(256, 4)
__global__ void expand_kernel(const float* __restrict__ coeffs,
                              const float* __restrict__ basis,
                              float* __restrict__ out) {
  const int tid  = threadIdx.x;
  const int wave = tid >> 5;
  const int lane = tid & 31;
  const int lh   = lane & 15;
  const int kh   = lane >> 4;

  const int f  = blockIdx.z;
  const int m0 = blockIdx.y * 16;                       // rows of (t*MB + b)
  const int n0 = blockIdx.x * 128 + wave * 16;          // pixel column base

  const float* A = coeffs + (size_t)f * (NTIME * MBATCH) * KD;  // [800][32]
  const float* B = basis  + (size_t)f * KD * NPIX;              // [32][16384]

  v2f Af[8];
#pragma unroll
  for (int kk = 0; kk < 8; ++kk) {
    int kb = kk * 4 + 2 * kh;
    Af[kk].x = A[(m0 + lh) * KD + kb];
    Af[kk].y = A[(m0 + lh) * KD + kb + 1];
  }

  v8f a0 = {}, a1 = {};
#pragma unroll
  for (int kk = 0; kk < 8; kk += 2) {
    int kb0 = kk * 4 + 2 * kh;
    v2f b0; b0.x = B[(size_t)kb0 * NPIX + n0 + lh];
            b0.y = B[(size_t)(kb0 + 1) * NPIX + n0 + lh];
    a0 = wmma4(Af[kk], b0, a0);
    int kb1 = (kk + 1) * 4 + 2 * kh;
    v2f b1v; b1v.x = B[(size_t)kb1 * NPIX + n0 + lh];
             b1v.y = B[(size_t)(kb1 + 1) * NPIX + n0 + lh];
    a1 = wmma4(Af[kk + 1], b1v, a1);
  }
  v8f acc = a0 + a1;

#pragma unroll
  for (int r = 0; r < 8; ++r) {
    int m = m0 + r + 8 * kh;                            // m = t*MB + b
    out[((size_t)m * 3 + f) * NPIX + n0 + lh] = acc[r];
  }
}

// ---------------------------------------------------------------------------
extern "C" void kernel_launch(void* const* d_in, const int* in_sizes, int n_in,
                              void* d_out, int out_size, void* d_ws, size_t ws_size,
                              hipStream_t stream) {
  // inputs: 0 grid0 (unused), 1 t, 2 init_coeffs, 3 W1, 4 b1, 5 W2, 6 b2,
  //         7 W3, 8 b3, 9 basis
  const float* t     = (const float*)d_in[1];
  const float* ic    = (const float*)d_in[2];
  const float* W1    = (const float*)d_in[3];
  const float* b1    = (const float*)d_in[4];
  const float* W2    = (const float*)d_in[5];
  const float* b2    = (const float*)d_in[6];
  const float* W3    = (const float*)d_in[7];
  const float* b3    = (const float*)d_in[8];
  const float* basis = (const float*)d_in[9];

  float* coeffs = (float*)d_ws;          // [3][100][8][32] = 307 KB scratch
  float* out    = (float*)d_out;         // [100][8][3][128][128]

  ode_rk4_kernel<<<3, 256, 0, stream>>>(t, ic, W1, b1, W2, b2, W3, b3, coeffs);
  expand_kernel<<<dim3(NPIX / 128, (NTIME * MBATCH) / 16, 3), 256, 0, stream>>>(
      coeffs, basis, out);
}